// Sim3_16492674417474
// MI455X (gfx1250) — compile-verified
//
#include <hip/hip_runtime.h>
#include <math.h>

typedef __attribute__((ext_vector_type(2))) float v2f;
typedef __attribute__((ext_vector_type(8))) float v8f;

#define TPp 4
#define TFf 2
#define NOFF 81
#define BN_EPS 1e-5f

// ---------------- GEMM (1x1 conv) + BN + SiLU, f32 WMMA 16x16x4 ----------------
// Y[d,p] = sum_c W[d,c] * X[c,p];  y = y*scale[d]+bias[d];  out = y*sigmoid(y)
// 8 waves = WM (M) x WNW (N) waves; each wave owns NSUB=2 pixel subtiles so one
// A fragment feeds two WMMAs per K-step.
template<int C, int HW, int DT>
__global__ __launch_bounds__(256) void conv_bn_silu(
    const float* __restrict__ x,   // (16, C, HW)
    const float* __restrict__ w,   // (C, C) row-major [d][c]
    const float* __restrict__ g, const float* __restrict__ bb,
    const float* __restrict__ m, const float* __restrict__ v,
    float* __restrict__ pf)        // (16, C, HW)
{
  constexpr int WM   = DT / 16;        // waves along M
  constexpr int WNW  = 8 / WM;         // waves along N
  constexpr int NSUB = 2;              // pixel subtiles per wave
  constexpr int PT   = WNW * NSUB * 16;

  __shared__ float Ws[DT * C];
  __shared__ float s_scale[DT];
  __shared__ float s_bias[DT];

  const int tid   = threadIdx.x;
  const int frame = blockIdx.z;
  const int dbase = blockIdx.y * DT;
  const int p0    = blockIdx.x * PT;

  // stage weight tile (coalesced along c)
  for (int i = tid; i < DT * C; i += 256) {
    int r = i / C, c = i % C;
    Ws[i] = w[(size_t)(dbase + r) * C + c];
  }
  if (tid < DT) {
    float sc = g[dbase + tid] * rsqrtf(v[dbase + tid] + BN_EPS);
    s_scale[tid] = sc;
    s_bias[tid]  = bb[dbase + tid] - m[dbase + tid] * sc;
  }
  __syncthreads();

  const int wave = tid >> 5;
  const int lane = tid & 31;
  const int wm   = wave / WNW;
  const int wn   = wave % WNW;
  const int mrow = wm * 16;          // local d base of this wave's tile
  const int half = lane >> 4;        // 0 | 1
  const int l16  = lane & 15;

  int p[NSUB], pc[NSUB];
#pragma unroll
  for (int s = 0; s < NSUB; ++s) {
    p[s]  = p0 + (wn * NSUB + s) * 16 + l16;
    pc[s] = (p[s] < HW) ? p[s] : (HW - 1);   // clamp loads; stores predicated
  }

  const float* xf = x + (size_t)frame * C * HW;

  v8f acc[NSUB] = {};
  for (int k = 0; k < C; k += 4) {
    const int ka = k + half * 2;
    // A (16x4): lane l16 -> M; VGPR j -> K = half*2 + j
    v2f a;
    a.x = Ws[(mrow + l16) * C + ka];
    a.y = Ws[(mrow + l16) * C + ka + 1];
    if ((k & 15) == 0 && (k + 16) < C)      // gfx1250 global_prefetch_b8
      __builtin_prefetch(&xf[(size_t)(ka + 16) * HW + pc[0]], 0, 3);
#pragma unroll
    for (int s = 0; s < NSUB; ++s) {
      // B (4x16): lane l16 -> N (pixel); VGPR j -> K = half*2 + j
      v2f b;
      b.x = xf[(size_t)ka * HW + pc[s]];
      b.y = xf[(size_t)(ka + 1) * HW + pc[s]];
      acc[s] = __builtin_amdgcn_wmma_f32_16x16x4_f32(false, a, false, b,
                                                     (short)0, acc[s], false, false);
    }
  }

  float* of = pf + (size_t)frame * C * HW;
#pragma unroll
  for (int s = 0; s < NSUB; ++s) {
    if (p[s] < HW) {
#pragma unroll
      for (int r = 0; r < 8; ++r) {
        const int dl = mrow + half * 8 + r;           // D layout: M = (lane>=16?8:0)+r
        float y = acc[s][r] * s_scale[dl] + s_bias[dl];
        float sg = 1.f / (1.f + __expf(-y));
        of[(size_t)(dbase + dl) * HW + p[s]] = y * sg;
      }
    }
  }
}

// ---------------- local correlation (81 offsets) + softmax + mlp dot ----------------
// VEC: widest LDS vector such that all shifted offsets stay aligned
// (STEP=4 -> float4, STEP=2 -> float2, STEP=1 -> scalar)
template<int C, int H, int W, int STEP, int VEC>
__global__ __launch_bounds__(256) void corr_softmax(
    const float* __restrict__ pf,   // (16, C, H*W) for this level
    const float* __restrict__ mlp,  // (81)
    const int* __restrict__ pci,    // (B, TP)
    const int* __restrict__ fci,    // (B, TF)
    float* __restrict__ out)        // (B, TF, C)
{
  constexpr int PAD = 4 * STEP;
  constexpr int WP  = W + 2 * PAD;
  constexpr int HP  = H + 2 * PAD;
  constexpr int HW  = H * W;
  constexpr int W4  = W / 4;

  extern __shared__ float smem[];
  float* f1p  = smem;                 // HP*WP padded f1 tile
  float* f2s  = f1p + HP * WP;        // HW
  float* sims = f2s + HW;             // 81
  __shared__ int s_t;

  const int tid = threadIdx.x;
  const int c   = blockIdx.x;
  const int n   = blockIdx.y;         // b*TF + tf
  const int b   = n / TFf;
  const int tf  = n % TFf;

  if (tid == 0) {
    // replicate _interp_indices scan for this (b, tf); p_pos persists over tf
    int p_pos = TPp - 2;
    int sel = 0;
    for (int i = 0; i <= tf; ++i) {
      int f = fci[b * TFf + i];
      while (p_pos >= 0) {
        if (p_pos == 0 || f < -pci[b * TPp + p_pos - 1]) { sel = p_pos; break; }
        --p_pos;
      }
    }
    s_t = sel;
  }
  // zero-fill padded tile with b128 stores
  for (int i = tid; i < (HP * WP) / 4; i += 256)
    *(float4*)&f1p[i * 4] = make_float4(0.f, 0.f, 0.f, 0.f);
  __syncthreads();

  const float* f1 = pf + ((size_t)(b * TPp + (TPp - 1)) * C + c) * HW;
  const float* f2 = pf + ((size_t)(b * TPp + s_t) * C + c) * HW;
  // vectorized fills (all index bases are multiples of 4 floats)
  for (int i = tid; i < HW / 4; i += 256) {
    int h = i / W4, w4 = i % W4;
    *(float4*)&f1p[(h + PAD) * WP + PAD + 4 * w4] = *(const float4*)&f1[h * W + 4 * w4];
    *(float4*)&f2s[i * 4] = *(const float4*)&f2[i * 4];
  }
  __syncthreads();

  const int wave = tid >> 5;
  const int lane = tid & 31;
  for (int o = wave; o < NOFF; o += 8) {
    const int oy = o / 9, ox = o % 9;
    const int rowoff = oy * STEP, coloff = ox * STEP;
    float acc = 0.f;
    if constexpr (VEC == 4) {
      for (int i = lane; i < HW / 4; i += 32) {
        int h = i / W4, w4 = i % W4;
        float4 a = *(const float4*)&f1p[(h + rowoff) * WP + 4 * w4 + coloff];
        float4 bq = *(const float4*)&f2s[i * 4];
        acc += a.x * bq.x + a.y * bq.y + a.z * bq.z + a.w * bq.w;
      }
    } else if constexpr (VEC == 2) {
      constexpr int W2 = W / 2;
      for (int i = lane; i < HW / 2; i += 32) {
        int h = i / W2, w2 = i % W2;
        float2 a = *(const float2*)&f1p[(h + rowoff) * WP + 2 * w2 + coloff];
        float2 bq = *(const float2*)&f2s[i * 2];
        acc += a.x * bq.x + a.y * bq.y;
      }
    } else {
      for (int i = lane; i < HW; i += 32) {
        int h = i / W, ww = i % W;
        acc += f1p[(h + rowoff) * WP + ww + coloff] * f2s[i];
      }
    }
#pragma unroll
    for (int msk = 16; msk >= 1; msk >>= 1) acc += __shfl_xor(acc, msk, 32);
    if (lane == 0) sims[o] = acc;
  }
  __syncthreads();

  if (tid == 0) {
    float mx = sims[0];
    for (int o = 1; o < NOFF; ++o) mx = fmaxf(mx, sims[o]);
    float sum = 0.f, dot = 0.f;
    for (int o = 0; o < NOFF; ++o) {
      float e = __expf(sims[o] - mx);
      sum += e;
      dot += e * mlp[o];
    }
    out[((size_t)b * TFf + tf) * C + c] = dot / sum;
  }
}

extern "C" void kernel_launch(void* const* d_in, const int* in_sizes, int n_in,
                              void* d_out, int out_size, void* d_ws, size_t ws_size,
                              hipStream_t stream) {
  (void)in_sizes; (void)n_in; (void)out_size; (void)ws_size;
  // setup_inputs() dict order:
  const float* f0 = (const float*)d_in[0];
  const float* w0 = (const float*)d_in[1];
  const float* g0 = (const float*)d_in[2];
  const float* b0 = (const float*)d_in[3];
  const float* m0 = (const float*)d_in[4];
  const float* v0 = (const float*)d_in[5];
  const float* f1 = (const float*)d_in[6];
  const float* w1 = (const float*)d_in[7];
  const float* g1 = (const float*)d_in[8];
  const float* b1 = (const float*)d_in[9];
  const float* m1 = (const float*)d_in[10];
  const float* v1 = (const float*)d_in[11];
  const float* f2 = (const float*)d_in[12];
  const float* w2 = (const float*)d_in[13];
  const float* g2 = (const float*)d_in[14];
  const float* b2 = (const float*)d_in[15];
  const float* m2 = (const float*)d_in[16];
  const float* v2 = (const float*)d_in[17];
  const float* mlp = (const float*)d_in[18];
  const int* pci   = (const int*)d_in[19];
  const int* fci   = (const int*)d_in[20];
  float* out = (float*)d_out;
  float* ws  = (float*)d_ws;

  // post-conv activations, all 16 frames per level, in workspace
  float* pf0 = ws;                                     // 16*128*6400
  float* pf1 = pf0 + (size_t)16 * 128 * 6400;          // 16*256*1600
  float* pf2 = pf1 + (size_t)16 * 256 * 1600;          // 16*512*400

  dim3 blk(256);
  // level 0: C=128, HW=6400, DT=32, PT=128 -> grid (6400/128, 128/32, 16)
  conv_bn_silu<128, 6400, 32><<<dim3(50, 4, 16), blk, 0, stream>>>(f0, w0, g0, b0, m0, v0, pf0);
  // level 1: C=256, HW=1600, DT=32, PT=128 -> grid (ceil(1600/128), 256/32, 16)
  conv_bn_silu<256, 1600, 32><<<dim3(13, 8, 16), blk, 0, stream>>>(f1, w1, g1, b1, m1, v1, pf1);
  // level 2: C=512, HW=400, DT=16, PT=256 -> grid (ceil(400/256), 512/16, 16)
  conv_bn_silu<512, 400, 16><<<dim3(2, 32, 16), blk, 0, stream>>>(f2, w2, g2, b2, m2, v2, pf2);

  const size_t sm0 = (size_t)(112 * 112 + 80 * 80 + NOFF) * sizeof(float);
  const size_t sm1 = (size_t)(56 * 56 + 40 * 40 + NOFF) * sizeof(float);
  const size_t sm2 = (size_t)(28 * 28 + 20 * 20 + NOFF) * sizeof(float);
  corr_softmax<128, 80, 80, 4, 4><<<dim3(128, 8), blk, sm0, stream>>>(pf0, mlp, pci, fci, out + 0);
  corr_softmax<256, 40, 40, 2, 2><<<dim3(256, 8), blk, sm1, stream>>>(pf1, mlp, pci, fci, out + 1024);
  corr_softmax<512, 20, 20, 1, 1><<<dim3(512, 8), blk, sm2, stream>>>(pf2, mlp, pci, fci, out + 3072);
}